// Memory_79551384256730
// MI455X (gfx1250) — compile-verified
//
#include <hip/hip_runtime.h>
#include <math.h>

#define BB 16
#define NN 2048
#define WW 64
#define RR 4
#define EPSF 1e-8f

typedef float v2f __attribute__((ext_vector_type(2)));
typedef float v8f __attribute__((ext_vector_type(8)));
typedef int v4i_ __attribute__((vector_size(16)));  // matches builtin param type

#if defined(__has_builtin)
#if __has_builtin(__builtin_amdgcn_global_load_async_to_lds_b128)
#define HAVE_ASYNC_COPY 1
#endif
#if __has_builtin(__builtin_amdgcn_s_wait_asynccnt)
#define HAVE_WAIT_ASYNC 1
#endif
#endif

__device__ __forceinline__ void wait_async0() {
#if defined(HAVE_WAIT_ASYNC)
  __builtin_amdgcn_s_wait_asynccnt(0);
#else
  asm volatile("s_wait_asynccnt 0x0" ::: "memory");
#endif
}

// ---------------------------------------------------------------------------
// K0: usage update, write-key cosine logits, link diagonal extraction
// ---------------------------------------------------------------------------
__global__ __launch_bounds__(256) void k0_prep(
    const float* __restrict__ mem, const float* __restrict__ usage_v,
    const float* __restrict__ wwp, const float* __restrict__ rwp,
    const float* __restrict__ wkey, const float* __restrict__ wstr,
    const float* __restrict__ fg, const float* __restrict__ link,
    float* __restrict__ ws_usage, float* __restrict__ ws_z,
    float* __restrict__ ws_diag) {
  int t = blockIdx.x * blockDim.x + threadIdx.x;
  if (t >= BB * NN) return;
  int b = t / NN, n = t % NN;
  float4 r4 = *(const float4*)(rwp + (size_t)t * 4);
  const float* fgb = fg + b * RR;
  float ret = (2.f - r4.x * fgb[0]) * (2.f - r4.y * fgb[1]) *
              (2.f - r4.z * fgb[2]) * (2.f - r4.w * fgb[3]);
  float u = usage_v[t], w0 = wwp[t];
  ws_usage[t] = (u + w0 - u * w0) * ret;

  const float* mrow = mem + (size_t)t * WW;
  const float* kb = wkey + b * WW;
  float dot = 0.f, nm = 0.f, nk = 0.f;
  for (int w = 0; w < WW; w += 4) {
    float4 m = *(const float4*)(mrow + w);
    float4 k = *(const float4*)(kb + w);
    dot += m.x * k.x + m.y * k.y + m.z * k.z + m.w * k.w;
    nm += m.x * m.x + m.y * m.y + m.z * m.z + m.w * m.w;
    nk += k.x * k.x + k.y * k.y + k.z * k.z + k.w * k.w;
  }
  float c = dot / (sqrtf(nm) * sqrtf(nk) + EPSF);
  ws_z[t] = c * wstr[b];
  ws_diag[t] = link[(size_t)b * NN * NN + (size_t)n * NN + n];
}

// ---------------------------------------------------------------------------
// K2: per-batch: softmax(write logits), stable bitonic sort of usage,
// exclusive cumprod, allocation scatter, write_w, WMMA B-operand matrix, P/Q
// ---------------------------------------------------------------------------
__global__ __launch_bounds__(1024) void k2_sortalloc(
    const float* __restrict__ ws_usage, const float* __restrict__ ws_z,
    const float* __restrict__ prec, const float* __restrict__ rwp,
    const float* __restrict__ wgate, const float* __restrict__ agate,
    float* __restrict__ ws_lookup, float* __restrict__ ws_ww,
    float* __restrict__ ws_Bmat, float* __restrict__ ws_scal) {
  __shared__ float sv[NN];
  __shared__ int si[NN];
  __shared__ float cp[NN];
  __shared__ float red[1024];
  int b = blockIdx.x, t = threadIdx.x;
  int n0 = t, n1 = t + 1024;
  const float* ub = ws_usage + b * NN;
  const float* zb = ws_z + b * NN;

  // softmax over z (content lookup for the write head)
  float z0 = zb[n0], z1 = zb[n1];
  red[t] = fmaxf(z0, z1);
  __syncthreads();
  for (int s = 512; s > 0; s >>= 1) { if (t < s) red[t] = fmaxf(red[t], red[t + s]); __syncthreads(); }
  float zmax = red[0];
  __syncthreads();
  float e0 = __expf(z0 - zmax), e1 = __expf(z1 - zmax);
  red[t] = e0 + e1;
  __syncthreads();
  for (int s = 512; s > 0; s >>= 1) { if (t < s) red[t] += red[t + s]; __syncthreads(); }
  float inv = 1.f / red[0];
  __syncthreads();
  float lw0 = e0 * inv, lw1 = e1 * inv;
  ws_lookup[b * NN + n0] = lw0;
  ws_lookup[b * NN + n1] = lw1;

  // bitonic sort ascending on (usage, index) -> emulates stable argsort
  sv[n0] = ub[n0]; si[n0] = n0;
  sv[n1] = ub[n1]; si[n1] = n1;
  __syncthreads();
  for (int k = 2; k <= NN; k <<= 1) {
    for (int j = k >> 1; j > 0; j >>= 1) {
      int i1 = ((t / j) * (2 * j)) + (t % j);
      int i2 = i1 + j;
      bool up = ((i1 & k) == 0);
      float v1 = sv[i1], v2 = sv[i2];
      int a1 = si[i1], a2 = si[i2];
      bool gt = (v1 > v2) || (v1 == v2 && a1 > a2);
      if (gt == up) { sv[i1] = v2; sv[i2] = v1; si[i1] = a2; si[i2] = a1; }
      __syncthreads();
    }
  }
  // exclusive cumprod of sorted usage (Hillis-Steele product scan, in place)
  cp[n0] = (n0 == 0) ? 1.f : sv[n0 - 1];
  cp[n1] = sv[n1 - 1];
  __syncthreads();
  for (int off = 1; off < NN; off <<= 1) {
    float t0 = (n0 >= off) ? cp[n0 - off] : 1.f;
    float t1 = (n1 >= off) ? cp[n1 - off] : 1.f;
    __syncthreads();
    cp[n0] *= t0; cp[n1] *= t1;
    __syncthreads();
  }
  float un0 = (1.f - sv[n0]) * cp[n0];
  float un1 = (1.f - sv[n1]) * cp[n1];
  __syncthreads();
  cp[si[n0]] = un0;  // scatter allocation weight back to original slot
  cp[si[n1]] = un1;
  __syncthreads();

  float wg = wgate[b], ag = agate[b];
  const float* rb = rwp + (size_t)b * NN * 4;
  const float* pb = prec + b * NN;
  float accP[4] = {0, 0, 0, 0}, accQ[4] = {0, 0, 0, 0};
  for (int pass = 0; pass < 2; ++pass) {
    int n = pass ? n1 : n0;
    float alloc = cp[n];
    float lw = pass ? lw1 : lw0;
    float wwv = wg * (ag * alloc + (1.f - ag) * lw);
    ws_ww[b * NN + n] = wwv;
    float4 r4 = *(const float4*)(rb + (size_t)n * 4);
    // WMMA B operand row: cols 0-3 = rwp, cols 4-7 = ww*rwp, 8-15 = 0
    float* bm = ws_Bmat + ((size_t)b * NN + n) * 16;
    bm[0] = r4.x; bm[1] = r4.y; bm[2] = r4.z; bm[3] = r4.w;
    bm[4] = wwv * r4.x; bm[5] = wwv * r4.y; bm[6] = wwv * r4.z; bm[7] = wwv * r4.w;
    for (int c = 8; c < 16; ++c) bm[c] = 0.f;
    float pv = pb[n];
    accP[0] += pv * r4.x; accP[1] += pv * r4.y; accP[2] += pv * r4.z; accP[3] += pv * r4.w;
    accQ[0] += wwv * r4.x; accQ[1] += wwv * r4.y; accQ[2] += wwv * r4.z; accQ[3] += wwv * r4.w;
  }
  for (int c = 0; c < 8; ++c) {
    float v = (c < 4) ? accP[c] : accQ[c - 4];
    __syncthreads();
    red[t] = v;
    __syncthreads();
    for (int s = 512; s > 0; s >>= 1) { if (t < s) red[t] += red[t + s]; __syncthreads(); }
    if (t == 0) ws_scal[b * 16 + c] = red[0];
  }
}

// ---------------------------------------------------------------------------
// K3: memory erase/write + read-key cosine logits (fused, one pass over mem)
// ---------------------------------------------------------------------------
__global__ __launch_bounds__(256) void k3_memupd(
    const float* __restrict__ mem, const float* __restrict__ ws_ww,
    const float* __restrict__ ev, const float* __restrict__ wv,
    const float* __restrict__ rkeys, const float* __restrict__ rstr,
    float* __restrict__ ws_mem, float* __restrict__ ws_zread) {
  __shared__ float sred[256];
  int b = blockIdx.y;
  int row = blockIdx.x * 4 + (threadIdx.x >> 6);
  int w = threadIdx.x & 63;
  int grp = threadIdx.x >> 6;
  size_t idx = ((size_t)b * NN + row) * WW + w;
  float wwv = ws_ww[b * NN + row];
  float mnew = mem[idx] * (1.f - wwv * ev[b * WW + w]) + wwv * wv[b * WW + w];
  ws_mem[idx] = mnew;
  float4 rk = *(const float4*)(rkeys + ((size_t)b * WW + w) * 4);
  float vals[9] = {mnew * rk.x, mnew * rk.y, mnew * rk.z, mnew * rk.w,
                   mnew * mnew, rk.x * rk.x, rk.y * rk.y, rk.z * rk.z, rk.w * rk.w};
  float out[9];
  for (int q = 0; q < 9; ++q) {
    sred[threadIdx.x] = vals[q];
    __syncthreads();
    for (int s = 32; s > 0; s >>= 1) {
      if (w < s) sred[threadIdx.x] += sred[threadIdx.x + s];
      __syncthreads();
    }
    out[q] = sred[grp << 6];
    __syncthreads();
  }
  if (w < RR) {
    float c = out[w] / (sqrtf(out[4]) * sqrtf(out[5 + w]) + EPSF);
    ws_zread[((size_t)b * NN + row) * 4 + w] = c * rstr[b * RR + w];
  }
}

// ---------------------------------------------------------------------------
// K4: per-batch softmax over N for the 4 read heads (in place on ws_zread)
// ---------------------------------------------------------------------------
__global__ __launch_bounds__(256) void k4_readsoftmax(float* __restrict__ ws_zread) {
  __shared__ float red[256];
  int b = blockIdx.x, t = threadIdx.x;
  float* zb = ws_zread + (size_t)b * NN * 4;
  float mx[4] = {-1e30f, -1e30f, -1e30f, -1e30f};
  for (int n = t; n < NN; n += 256)
    for (int r = 0; r < 4; ++r) mx[r] = fmaxf(mx[r], zb[n * 4 + r]);
  float M4[4];
  for (int r = 0; r < 4; ++r) {
    red[t] = mx[r];
    __syncthreads();
    for (int s = 128; s > 0; s >>= 1) { if (t < s) red[t] = fmaxf(red[t], red[t + s]); __syncthreads(); }
    M4[r] = red[0];
    __syncthreads();
  }
  float sm[4] = {0, 0, 0, 0};
  for (int n = t; n < NN; n += 256)
    for (int r = 0; r < 4; ++r) sm[r] += __expf(zb[n * 4 + r] - M4[r]);
  float I4[4];
  for (int r = 0; r < 4; ++r) {
    red[t] = sm[r];
    __syncthreads();
    for (int s = 128; s > 0; s >>= 1) { if (t < s) red[t] += red[t + s]; __syncthreads(); }
    I4[r] = 1.f / red[0];
    __syncthreads();
  }
  for (int n = t; n < NN; n += 256)
    for (int r = 0; r < 4; ++r) zb[n * 4 + r] = __expf(zb[n * 4 + r] - M4[r]) * I4[r];
}

// ---------------------------------------------------------------------------
// K5: dominant pass over the 268 MB link matrix (read exactly once).
// One wave per (batch, 128-row strip, 512-col chunk). WMMA f32 16x16x4
// computes S1/S2 (row direction) and T1/T2 (transposed direction) against a
// packed 16-col B operand. Tiles are staged into LDS with CDNA5 async
// global->LDS copies (ASYNCcnt-tracked), double-buffered so the prefetch of
// tile jt+1 overlaps the 64 WMMAs of tile jt. Pitch 20 floats gives
// 16B-aligned rows and conflict-free transposed (column) ds reads.
// ---------------------------------------------------------------------------
#define LP 20  // LDS tile pitch in floats

__global__ __launch_bounds__(32) void k5_link(
    const float* __restrict__ link, const float* __restrict__ ws_Bmat,
    float* __restrict__ ws_Spart, float* __restrict__ ws_Tpart) {
  __shared__ float lds[2][128 * LP];
  const int lane = threadIdx.x;
  const int jc = blockIdx.x;     // 0..3   (512-wide column chunk)
  const int strip = blockIdx.y;  // 0..15  (128-row strip)
  const int b = blockIdx.z;      // 0..15
  const int i0 = strip * 128;
  const int jbase = jc * 512;
  const float* L = link + (size_t)b * NN * NN;
  const float* Bm = ws_Bmat + (size_t)b * NN * 16;

  v8f zero = {0, 0, 0, 0, 0, 0, 0, 0};
  v8f Sacc[8];
  for (int rt = 0; rt < 8; ++rt) Sacc[rt] = zero;

  const int lr = lane >> 2, lc = (lane & 3) * 4;
  const int mrow = lane & 15;
  const int khalf = (lane < 16) ? 0 : 2;

  // stage one 128x16 tile into LDS buffer `buf`
  auto stage = [&](int jt, int buf) {
    int j = jbase + jt * 16;
#if defined(HAVE_ASYNC_COPY)
    for (int it = 0; it < 16; ++it) {
      int rloc = it * 8 + lr;
      const float* g = L + (size_t)(i0 + rloc) * NN + j + lc;
      float* l = &lds[buf][rloc * LP + lc];
      __builtin_amdgcn_global_load_async_to_lds_b128(
          (__attribute__((address_space(1))) v4i_*)g,
          (__attribute__((address_space(3))) v4i_*)l,
          /*offset=*/0, /*cpol=*/0);
    }
#else
    for (int it = 0; it < 16; ++it) {
      int rloc = it * 8 + lr;
      float4 v = *(const float4*)(L + (size_t)(i0 + rloc) * NN + j + lc);
      *(float4*)(&lds[buf][rloc * LP + lc]) = v;
    }
#endif
  };

  stage(0, 0);
  for (int jt = 0; jt < 32; ++jt) {
    int j = jbase + jt * 16;
    int cur = jt & 1;
#if defined(HAVE_ASYNC_COPY)
    wait_async0();  // tile jt resident in lds[cur]
#else
    __syncthreads();
#endif
    if (jt + 1 < 32) stage(jt + 1, cur ^ 1);  // overlap prefetch with compute
    const float* tile = lds[cur];

    // B fragments for S direction (rows j..j+15 of Bmat), reused over row tiles
    v2f bS[4];
    for (int k = 0; k < 4; ++k) {
      int rowb = (j + k * 4) * 16;
      bS[k].x = Bm[rowb + lane];
      bS[k].y = Bm[rowb + 32 + lane];
    }
    // S: 8 row tiles x 4 k-steps, K = j
    for (int rt = 0; rt < 8; ++rt) {
      for (int k = 0; k < 4; ++k) {
        int koff = k * 4 + khalf;
        v2f a;
        a.x = tile[(rt * 16 + mrow) * LP + koff];
        a.y = tile[(rt * 16 + mrow) * LP + koff + 1];
        Sacc[rt] = __builtin_amdgcn_wmma_f32_16x16x4_f32(
            false, a, false, bS[k], (short)0, Sacc[rt], false, false);
      }
    }
    // T: transposed tile, M = j-tile, K = local rows (128 -> 32 k-steps)
    v8f Tacc = zero;
    for (int kk = 0; kk < 32; ++kk) {
      int koff = kk * 4 + khalf;
      v2f a;  // A[M=jm, K] = L[K, j+jm] : row reads of the LDS tile
      a.x = tile[koff * LP + mrow];
      a.y = tile[(koff + 1) * LP + mrow];
      v2f bT;
      int rowb = (i0 + kk * 4) * 16;
      bT.x = Bm[rowb + lane];
      bT.y = Bm[rowb + 32 + lane];
      Tacc = __builtin_amdgcn_wmma_f32_16x16x4_f32(
          false, a, false, bT, (short)0, Tacc, false, false);
    }
    // flush transposed partial: Tpart[b][strip][j..j+15][0..15]
    float* Tout = ws_Tpart + (((size_t)b * 16 + strip) * NN + j) * 16;
    int col = lane & 15;
    int mofs = (lane < 16) ? 0 : 8;
    for (int v = 0; v < 8; ++v) Tout[(v + mofs) * 16 + col] = Tacc[v];
#if !defined(HAVE_ASYNC_COPY)
    __syncthreads();
#endif
  }
  // flush S partials: Spart[b][jc][i0..i0+127][0..15]
  float* Sout = ws_Spart + (((size_t)b * 4 + jc) * NN + i0) * 16;
  int col = lane & 15;
  int mofs = (lane < 16) ? 0 : 8;
  for (int rt = 0; rt < 8; ++rt)
    for (int v = 0; v < 8; ++v)
      Sout[(rt * 16 + v + mofs) * 16 + col] = Sacc[rt][v];
}

// ---------------------------------------------------------------------------
// K6: deterministic reduction of S (over 4 col-chunks) and T (over 16 strips)
// ---------------------------------------------------------------------------
__global__ __launch_bounds__(256) void k6_reduce(
    const float* __restrict__ Spart, const float* __restrict__ Tpart,
    float* __restrict__ Sred, float* __restrict__ Tred) {
  int t = blockIdx.x * blockDim.x + threadIdx.x;  // B*N*8
  if (t >= BB * NN * 8) return;
  int c = t & 7;
  int n = (t >> 3) % NN;
  int b = t / (NN * 8);
  float s = 0.f;
  for (int jc = 0; jc < 4; ++jc) s += Spart[(((size_t)b * 4 + jc) * NN + n) * 16 + c];
  Sred[t] = s;
  float tv = 0.f;
  for (int st = 0; st < 16; ++st) tv += Tpart[(((size_t)b * 16 + st) * NN + n) * 16 + c];
  Tred[t] = tv;
}

// ---------------------------------------------------------------------------
// K7: fwd/bwd via rank-1 + diagonal corrections, combine with read modes,
// emit padded rw matrix (N x 16, cols 0-3 live) for the final WMMA GEMM
// ---------------------------------------------------------------------------
__global__ __launch_bounds__(256) void k7_combine(
    const float* __restrict__ Sred, const float* __restrict__ Tred,
    const float* __restrict__ read_lookup, const float* __restrict__ ws_ww,
    const float* __restrict__ ws_diag, const float* __restrict__ prec,
    const float* __restrict__ rwp, const float* __restrict__ ws_scal,
    const float* __restrict__ modes, float* __restrict__ ws_rw16) {
  int t = blockIdx.x * blockDim.x + threadIdx.x;
  if (t >= BB * NN) return;
  int b = t / NN;
  const float* S = Sred + (size_t)t * 8;
  const float* T = Tred + (size_t)t * 8;
  float wwv = ws_ww[t], Lii = ws_diag[t], pv = prec[t];
  float4 r4 = *(const float4*)(rwp + (size_t)t * 4);
  float4 rl = *(const float4*)(read_lookup + (size_t)t * 4);
  const float* sc = ws_scal + b * 16;  // P[0..3], Q[4..7]
  const float* mo = modes + b * 12;    // (3, R)
  float rr[4] = {r4.x, r4.y, r4.z, r4.w};
  float rlv[4] = {rl.x, rl.y, rl.z, rl.w};
  float* out = ws_rw16 + (size_t)t * 16;
  for (int r = 0; r < 4; ++r) {
    float diagc = (1.f - 2.f * wwv) * Lii * rr[r] + wwv * pv * rr[r];
    float fwd = (1.f - wwv) * S[r] - S[4 + r] + wwv * sc[r] - diagc;
    float bwd = (1.f - wwv) * T[r] - T[4 + r] + pv * sc[4 + r] - diagc;
    out[r] = mo[0 * 4 + r] * bwd + mo[1 * 4 + r] * rlv[r] + mo[2 * 4 + r] * fwd;
  }
  for (int c = 4; c < 16; ++c) out[c] = 0.f;
}

// ---------------------------------------------------------------------------
// K8: read_vecs[b,w,r] = sum_n mem_new[b,n,w]*rw[b,n,r] via WMMA (M=w tiles,
// K=n, N'=r padded to 16). Deterministic: each wave owns full K, no atomics.
// ---------------------------------------------------------------------------
__global__ __launch_bounds__(32) void k8_readvecs(
    const float* __restrict__ ws_mem, const float* __restrict__ ws_rw16,
    float* __restrict__ out) {
  const int lane = threadIdx.x;
  const int wt = blockIdx.x;  // 0..3 (16-wide w tile)
  const int b = blockIdx.y;   // 0..15
  const int w0 = wt * 16;
  const float* M = ws_mem + (size_t)b * NN * WW;
  const float* Rw = ws_rw16 + (size_t)b * NN * 16;
  v8f acc = {0, 0, 0, 0, 0, 0, 0, 0};
  const int wm = lane & 15;
  const int khalf = (lane < 16) ? 0 : 2;
  for (int kk = 0; kk < NN / 4; ++kk) {
    int k0 = kk * 4;
    int koff = k0 + khalf;
    v2f a;  // A[M=w, K=n] = mem[n, w]  (transposed read, coalesced by lane)
    a.x = M[(size_t)koff * WW + w0 + wm];
    a.y = M[(size_t)(koff + 1) * WW + w0 + wm];
    v2f bf;
    bf.x = Rw[(size_t)k0 * 16 + lane];
    bf.y = Rw[(size_t)k0 * 16 + 32 + lane];
    acc = __builtin_amdgcn_wmma_f32_16x16x4_f32(
        false, a, false, bf, (short)0, acc, false, false);
  }
  int col = lane & 15;
  int mofs = (lane < 16) ? 0 : 8;
  if (col < RR) {
    for (int v = 0; v < 8; ++v) {
      int w = w0 + v + mofs;
      out[((size_t)b * WW + w) * RR + col] = acc[v];
    }
  }
}

// ---------------------------------------------------------------------------
extern "C" void kernel_launch(void* const* d_in, const int* in_sizes, int n_in,
                              void* d_out, int out_size, void* d_ws, size_t ws_size,
                              hipStream_t stream) {
  const float* mem   = (const float*)d_in[0];
  const float* usage = (const float*)d_in[1];
  const float* prec  = (const float*)d_in[2];
  const float* link  = (const float*)d_in[3];
  const float* wwp   = (const float*)d_in[4];
  const float* rwp   = (const float*)d_in[5];
  const float* wkey  = (const float*)d_in[6];
  const float* wstr  = (const float*)d_in[7];
  const float* fg    = (const float*)d_in[8];
  const float* wg    = (const float*)d_in[9];
  const float* ag    = (const float*)d_in[10];
  const float* wv    = (const float*)d_in[11];
  const float* ev    = (const float*)d_in[12];
  const float* rkeys = (const float*)d_in[13];
  const float* rstr  = (const float*)d_in[14];
  const float* modes = (const float*)d_in[15];
  float* out = (float*)d_out;

  float* ws = (float*)d_ws;
  size_t off = 0;
  float* ws_usage  = ws + off; off += (size_t)BB * NN;
  float* ws_z      = ws + off; off += (size_t)BB * NN;
  float* ws_lookup = ws + off; off += (size_t)BB * NN;
  float* ws_ww     = ws + off; off += (size_t)BB * NN;
  float* ws_diag   = ws + off; off += (size_t)BB * NN;
  float* ws_Bmat   = ws + off; off += (size_t)BB * NN * 16;
  float* ws_scal   = ws + off; off += (size_t)BB * 16;
  float* ws_mem    = ws + off; off += (size_t)BB * NN * WW;
  float* ws_zread  = ws + off; off += (size_t)BB * NN * 4;
  float* ws_Spart  = ws + off; off += (size_t)BB * 4 * NN * 16;
  float* ws_Tpart  = ws + off; off += (size_t)BB * 16 * NN * 16;
  float* ws_Sred   = ws + off; off += (size_t)BB * NN * 8;
  float* ws_Tred   = ws + off; off += (size_t)BB * NN * 8;
  float* ws_rw16   = ws + off; off += (size_t)BB * NN * 16;
  (void)ws_size; (void)in_sizes; (void)n_in; (void)out_size;

  (void)hipMemsetAsync(d_out, 0, (size_t)out_size * sizeof(float), stream);

  k0_prep<<<(BB * NN + 255) / 256, 256, 0, stream>>>(
      mem, usage, wwp, rwp, wkey, wstr, fg, link, ws_usage, ws_z, ws_diag);
  k2_sortalloc<<<BB, 1024, 0, stream>>>(
      ws_usage, ws_z, prec, rwp, wg, ag, ws_lookup, ws_ww, ws_Bmat, ws_scal);
  k3_memupd<<<dim3(NN / 4, BB), 256, 0, stream>>>(
      mem, ws_ww, ev, wv, rkeys, rstr, ws_mem, ws_zread);
  k4_readsoftmax<<<BB, 256, 0, stream>>>(ws_zread);
  k5_link<<<dim3(4, 16, BB), 32, 0, stream>>>(link, ws_Bmat, ws_Spart, ws_Tpart);
  k6_reduce<<<(BB * NN * 8 + 255) / 256, 256, 0, stream>>>(
      ws_Spart, ws_Tpart, ws_Sred, ws_Tred);
  k7_combine<<<(BB * NN + 255) / 256, 256, 0, stream>>>(
      ws_Sred, ws_Tred, ws_zread, ws_ww, ws_diag, prec, rwp, ws_scal, modes, ws_rw16);
  k8_readvecs<<<dim3(4, BB), 32, 0, stream>>>(ws_mem, ws_rw16, out);
}